// HausdorffERLoss_40613210751066
// MI455X (gfx1250) — compile-verified
//
#include <hip/hip_runtime.h>
#include <stdint.h>

// ---------------------------------------------------------------------------
// HausdorffER loss, MI455X (gfx1250).
// 64 images of 512x512 fp32, 10 sequential erosion steps, scalar output.
// Memory-bound: ping-pong buffers (2 x 64MB) are L2-resident (192MB L2).
// CDNA5 paths: GLOBAL_LOAD_ASYNC_TO_LDS_B128/B32 halo staging (ASYNCcnt),
// s_wait_asynccnt, wave32 shfl reductions.
// ---------------------------------------------------------------------------

#define IMGS       64
#define HH         512
#define WW         512
#define NPIX       (HH * WW)          // 262144
#define TW         64
#define TH         16
#define TILES_X    (WW / TW)          // 8
#define TILES_Y    (HH / TH)          // 32
#define TILES_PER_IMG (TILES_X * TILES_Y)  // 256
#define LDS_H      (TH + 2)           // 18 halo rows
#define LDS_STRIDE 72                 // floats per row (288 B, 16B-multiple)
#define COL0       4                  // interior starts at col 4 (16B aligned)
#define NSTEPS     10

typedef int v4i __attribute__((ext_vector_type(4)));
typedef __attribute__((address_space(1))) int as1_int;
typedef __attribute__((address_space(3))) int as3_int;
typedef __attribute__((address_space(1))) v4i as1_v4i;
typedef __attribute__((address_space(3))) v4i as3_v4i;

// Async global -> LDS copies (per-lane). ASYNCcnt-tracked.
__device__ __forceinline__ void async_load_b32(const float* gsrc, float* ldst) {
#if __has_builtin(__builtin_amdgcn_global_load_async_to_lds_b32)
  __builtin_amdgcn_global_load_async_to_lds_b32(
      (as1_int*)(uintptr_t)gsrc,
      (as3_int*)(unsigned)(uintptr_t)ldst, 0, 0);
#else
  unsigned lds_addr = (unsigned)(uintptr_t)ldst;
  unsigned long long gad = (unsigned long long)(uintptr_t)gsrc;
  asm volatile("global_load_async_to_lds_b32 %0, %1, off"
               :: "v"(lds_addr), "v"(gad) : "memory");
#endif
}

__device__ __forceinline__ void async_load_b128(const float* gsrc, float* ldst) {
#if __has_builtin(__builtin_amdgcn_global_load_async_to_lds_b128)
  __builtin_amdgcn_global_load_async_to_lds_b128(
      (as1_v4i*)(uintptr_t)gsrc,
      (as3_v4i*)(unsigned)(uintptr_t)ldst, 0, 0);
#else
  async_load_b32(gsrc + 0, ldst + 0);
  async_load_b32(gsrc + 1, ldst + 1);
  async_load_b32(gsrc + 2, ldst + 2);
  async_load_b32(gsrc + 3, ldst + 3);
#endif
}

__device__ __forceinline__ void wait_async0() {
#if __has_builtin(__builtin_amdgcn_s_wait_asynccnt)
  __builtin_amdgcn_s_wait_asynccnt(0);
#else
  asm volatile("s_wait_asynccnt 0x0" ::: "memory");
#endif
}

// ---------------------------------------------------------------------------
// Erosion step on a 64x16 tile (placed first so disasm snippet shows it).
//   cur = (raw - mn) * inv  (affine applied on read; OOB halo filled with mn
//   so it maps to 0, matching zero padding in the normalized domain)
//   dil = 0.2*(c+n+s+e+w) of cur  ->  0.2*inv*S5 - mn*inv   (5*0.2 == 1)
//   ero = max(dil - 0.5, 0)       -> stored raw for the next step
// Per-workgroup partial (min, max, sum) of ero written to `partials`.
// ---------------------------------------------------------------------------
__global__ void __launch_bounds__(256)
stencil_kernel(const float* __restrict__ src, float* __restrict__ dst,
               const float* __restrict__ affine, float* __restrict__ partials) {
  __shared__ float tile[LDS_H * LDS_STRIDE];
  __shared__ float redMin[8], redMax[8], redSum[8];

  const int tid = threadIdx.x;
  const int img = blockIdx.z;
  const int x0  = blockIdx.x * TW;
  const int y0  = blockIdx.y * TH;

  const float mn  = affine[2 * img + 0];
  const float inv = affine[2 * img + 1];
  const float* simg = src + ((size_t)img << 18);

  // --- Stage halo tile ----------------------------------------------------
  // Interior: 18 rows x 16 segments of 16B -> async b128 (288 transfers).
  for (int i = tid; i < LDS_H * 16; i += 256) {
    int row = i >> 4;            // 0..17
    int seg = i & 15;            // 0..15
    int gy  = y0 + row - 1;
    float* lp = &tile[row * LDS_STRIDE + COL0 + seg * 4];
    if ((unsigned)gy < (unsigned)HH) {
      async_load_b128(&simg[(gy << 9) + x0 + seg * 4], lp);
    } else {
      *(float4*)lp = make_float4(mn, mn, mn, mn);   // OOB row -> mn
    }
  }
  // Halo columns: 18 rows x 2 sides -> async b32 (36 transfers).
  if (tid < LDS_H * 2) {
    int row  = tid >> 1;
    int side = tid & 1;
    int gy   = y0 + row - 1;
    int gx   = side ? (x0 + TW) : (x0 - 1);
    int lcol = side ? (COL0 + TW) : (COL0 - 1);
    float* lp = &tile[row * LDS_STRIDE + lcol];
    if ((unsigned)gy < (unsigned)HH && (unsigned)gx < (unsigned)WW) {
      async_load_b32(&simg[(gy << 9) + gx], lp);
    } else {
      *lp = mn;                                     // OOB corner/edge -> mn
    }
  }
  wait_async0();        // drain this wave's ASYNCcnt
  __syncthreads();      // DS stores + all waves' async writes now visible

  // --- Stencil + thread-local reduction -----------------------------------
  const int   c     = tid & (TW - 1);
  const int   r0    = tid >> 6;           // 0..3, each thread does rows r0+4j
  const float scale = 0.2f * inv;
  const float bias  = mn * inv;

  float lmin = 3.4e38f, lmax = -3.4e38f, lsum = 0.0f;
  float* dimg = dst + ((size_t)img << 18);

#pragma unroll
  for (int j = 0; j < 4; ++j) {
    int r = r0 + j * 4;
    const float* t = &tile[(r + 1) * LDS_STRIDE + (c + COL0)];
    float s5  = t[0] + t[-1] + t[1] + t[-LDS_STRIDE] + t[LDS_STRIDE];
    float dil = scale * s5 - bias;
    float ero = fmaxf(dil - 0.5f, 0.0f);
    dimg[((y0 + r) << 9) + (x0 + c)] = ero;
    lmin = fminf(lmin, ero);
    lmax = fmaxf(lmax, ero);
    lsum += ero;
  }

  // --- wave32 butterfly + cross-wave reduction ----------------------------
#pragma unroll
  for (int o = 16; o > 0; o >>= 1) {
    lmin = fminf(lmin, __shfl_xor(lmin, o, 32));
    lmax = fmaxf(lmax, __shfl_xor(lmax, o, 32));
    lsum += __shfl_xor(lsum, o, 32);
  }
  const int wave = tid >> 5;
  if ((tid & 31) == 0) { redMin[wave] = lmin; redMax[wave] = lmax; redSum[wave] = lsum; }
  __syncthreads();

  if (tid == 0) {
    float m0 = redMin[0], m1 = redMax[0], s = redSum[0];
#pragma unroll
    for (int w = 1; w < 8; ++w) {
      m0 = fminf(m0, redMin[w]);
      m1 = fmaxf(m1, redMax[w]);
      s += redSum[w];
    }
    const int tileIdx = blockIdx.y * TILES_X + blockIdx.x;
    float* p = partials + (size_t)(img * TILES_PER_IMG + tileIdx) * 3;
    p[0] = m0; p[1] = m1; p[2] = s;
  }
}

// ---------------------------------------------------------------------------
// Per-image affine = (0,1), running contributions = 0.
// ---------------------------------------------------------------------------
__global__ void init_kernel(float* __restrict__ affine, float* __restrict__ contrib) {
  int i = threadIdx.x;
  if (i < IMGS) {
    affine[2 * i + 0] = 0.0f;
    affine[2 * i + 1] = 1.0f;
    contrib[i] = 0.0f;
  }
}

// ---------------------------------------------------------------------------
// bound = (x - y)^2, vectorized float4.
// ---------------------------------------------------------------------------
__global__ void __launch_bounds__(256)
bound_kernel(const float4* __restrict__ x, const float4* __restrict__ y,
             float4* __restrict__ out, int n4) {
  int i = blockIdx.x * 256 + threadIdx.x;
  if (i < n4) {
    float4 a = x[i], b = y[i], r;
    r.x = (a.x - b.x) * (a.x - b.x);
    r.y = (a.y - b.y) * (a.y - b.y);
    r.z = (a.z - b.z) * (a.z - b.z);
    r.w = (a.w - b.w) * (a.w - b.w);
    out[i] = r;
  }
}

// ---------------------------------------------------------------------------
// Per-image finalize. Reduce 256 tile partials -> (mn, mx, sum);
// update affine for the next step and accumulate weighted normalized sum.
//   ptp > 0: affine = (mn, 1/ptp), contrib += w * (sum - N*mn)/ptp
//   ptp == 0: affine = (0, 1),     contrib += w * sum
// ---------------------------------------------------------------------------
__global__ void __launch_bounds__(256)
finalize_kernel(const float* __restrict__ partials, float* __restrict__ affine,
                float* __restrict__ contrib, float w) {
  __shared__ float redMin[8], redMax[8], redSum[8];
  const int img = blockIdx.x;
  const int tid = threadIdx.x;

  const float* p = partials + (size_t)(img * TILES_PER_IMG + tid) * 3;
  float lmin = p[0], lmax = p[1], lsum = p[2];

#pragma unroll
  for (int o = 16; o > 0; o >>= 1) {
    lmin = fminf(lmin, __shfl_xor(lmin, o, 32));
    lmax = fmaxf(lmax, __shfl_xor(lmax, o, 32));
    lsum += __shfl_xor(lsum, o, 32);
  }
  const int wave = tid >> 5;
  if ((tid & 31) == 0) { redMin[wave] = lmin; redMax[wave] = lmax; redSum[wave] = lsum; }
  __syncthreads();

  if (tid == 0) {
    float mn = redMin[0], mx = redMax[0], s = redSum[0];
#pragma unroll
    for (int v = 1; v < 8; ++v) {
      mn = fminf(mn, redMin[v]);
      mx = fmaxf(mx, redMax[v]);
      s += redSum[v];
    }
    float ptp = mx - mn;
    float c, nmn, ninv;
    if (ptp > 0.0f) {
      ninv = 1.0f / ptp;
      nmn  = mn;
      c    = (s - (float)NPIX * mn) * ninv;
    } else {
      ninv = 1.0f;
      nmn  = 0.0f;
      c    = s;
    }
    affine[2 * img + 0] = nmn;
    affine[2 * img + 1] = ninv;
    contrib[img] += w * c;
  }
}

// ---------------------------------------------------------------------------
// Final scalar = sum(contrib) / (64 * 262144). Fixed-order sum.
// ---------------------------------------------------------------------------
__global__ void final_kernel(const float* __restrict__ contrib, float* __restrict__ out) {
  if (threadIdx.x == 0) {
    float s = 0.0f;
    for (int i = 0; i < IMGS; ++i) s += contrib[i];
    out[0] = s / ((float)IMGS * (float)NPIX);
  }
}

// ---------------------------------------------------------------------------
extern "C" void kernel_launch(void* const* d_in, const int* in_sizes, int n_in,
                              void* d_out, int out_size, void* d_ws, size_t ws_size,
                              hipStream_t stream) {
  const float* x = (const float*)d_in[0];
  const float* y = (const float*)d_in[1];

  // Workspace layout (floats): bufA | bufB | partials | affine | contrib
  float* bufA     = (float*)d_ws;
  float* bufB     = bufA + (size_t)IMGS * NPIX;
  float* partials = bufB + (size_t)IMGS * NPIX;
  float* affine   = partials + (size_t)IMGS * TILES_PER_IMG * 3;
  float* contrib  = affine + (size_t)IMGS * 2;

  hipLaunchKernelGGL(init_kernel, dim3(1), dim3(64), 0, stream, affine, contrib);

  const int n4 = (IMGS * NPIX) / 4;
  hipLaunchKernelGGL(bound_kernel, dim3((n4 + 255) / 256), dim3(256), 0, stream,
                     (const float4*)x, (const float4*)y, (float4*)bufA, n4);

  dim3 sgrid(TILES_X, TILES_Y, IMGS);
  for (int k = 0; k < NSTEPS; ++k) {
    const float* src = (k & 1) ? bufB : bufA;
    float*       dst = (k & 1) ? bufA : bufB;
    hipLaunchKernelGGL(stencil_kernel, sgrid, dim3(256), 0, stream,
                       src, dst, affine, partials);
    float w = (float)((k + 1) * (k + 1));
    hipLaunchKernelGGL(finalize_kernel, dim3(IMGS), dim3(256), 0, stream,
                       partials, affine, contrib, w);
  }

  hipLaunchKernelGGL(final_kernel, dim3(1), dim3(64), 0, stream, contrib, (float*)d_out);
}